// AnchorTriangleAttention_7078106104182
// MI455X (gfx1250) — compile-verified
//
#include <hip/hip_runtime.h>
#include <hip/hip_bf16.h>

#define L 512
#define DIM 128
#define ADIM 64
#define KANCH 32

typedef float v2f __attribute__((ext_vector_type(2)));
typedef float v8f __attribute__((ext_vector_type(8)));

// ---- WMMA helpers: V_WMMA_F32_16X16X4_F32 (fp32, keeps reference precision) ----

__device__ __forceinline__ v8f wmma4(v2f a, v2f b, v8f c) {
  // 8 args: (neg_a, A, neg_b, B, c_mod, C, reuse_a, reuse_b)
  return __builtin_amdgcn_wmma_f32_16x16x4_f32(false, a, false, b, (short)0, c,
                                               false, false);
}

// A fragment: 16x4 (MxK) from row-major matrix with leading dim lda, k-base kb.
// Lane l: M = l&15 ; VGPR0 holds K = (l>>4)*2, VGPR1 holds K = (l>>4)*2 + 1.
__device__ __forceinline__ v2f fragA(const float* A, int lda, int kb, int lane) {
  int m = lane & 15;
  int kq = (lane >> 4) << 1;
  v2f a;
  a.x = A[m * lda + kb + kq];
  a.y = A[m * lda + kb + kq + 1];
  return a;
}

// B fragment: 4x16 (KxN) from row-major KxN matrix (ldb = N stride), n-base nb.
__device__ __forceinline__ v2f fragB(const float* B, int ldb, int kb, int nb, int lane) {
  int n = lane & 15;
  int kq = (lane >> 4) << 1;
  v2f b;
  b.x = B[(kb + kq) * ldb + nb + n];
  b.y = B[(kb + kq + 1) * ldb + nb + n];
  return b;
}

// B fragment where B = transpose of a row-major N x K matrix M (B[k][n] = M[n][k]).
__device__ __forceinline__ v2f fragBT(const float* M, int ldm, int kb, int nb, int lane) {
  int n = lane & 15;
  int kq = (lane >> 4) << 1;
  v2f b;
  b.x = M[(nb + n) * ldm + kb + kq];
  b.y = M[(nb + n) * ldm + kb + kq + 1];
  return b;
}

// D store: 16x16 f32, VGPR r holds M=r (lanes 0-15) / M=r+8 (lanes 16-31).
__device__ __forceinline__ void storeD(float* O, int ldo, const v8f& d, int lane) {
  int n = lane & 15;
  int mb = (lane >> 4) << 3;
#pragma unroll
  for (int r = 0; r < 8; ++r) O[(mb + r) * ldo + n] = d[r];
}

// ---------------- Kernel 0: template gate scalar ----------------

__global__ __launch_bounds__(256) void gate_kernel(
    const float* __restrict__ td, const float* __restrict__ quality,
    const float* __restrict__ W1, const float* __restrict__ b1,
    const float* __restrict__ W2, const float* __restrict__ b2,
    float* __restrict__ gate_out) {
  __shared__ float red[256];
  int tid = threadIdx.x;
  float s = 0.f;
  for (int i = tid; i < L * L; i += 256) s += (td[i] > 0.f) ? 1.f : 0.f;
  red[tid] = s;
  __syncthreads();
  for (int w = 128; w > 0; w >>= 1) {
    if (tid < w) red[tid] += red[tid + w];
    __syncthreads();
  }
  if (tid == 0) {
    float coverage = red[0] / (float)(L * L);
    float f0 = coverage, f1 = quality[0], f2 = 1.0f;  // L/512
    float acc = b2[0];
#pragma unroll
    for (int h = 0; h < 16; ++h) {
      float hv = f0 * W1[0 * 16 + h] + f1 * W1[1 * 16 + h] + f2 * W1[2 * 16 + h] + b1[h];
      hv = hv > 0.f ? hv : 0.f;
      acc += hv * W2[h];
    }
    gate_out[0] = 1.f / (1.f + __expf(-acc));
  }
}

// -------- Kernel 1: anchor gathers + projections (left/right/v_left/v_right) --------
// side 0: rows r = i*K + k, X[r,:] = pair[i, anchor[k], :] -> left, v_left
// side 1: rows r = k*L + j, X[r,:] = pair[anchor[k], j, :] -> right, v_right
// One wave per 16-row tile; 256 WMMAs per block.

__global__ __launch_bounds__(32) void anchor_proj_kernel(
    const float* __restrict__ pair, const int* __restrict__ anchor,
    const float* __restrict__ Wl, const float* __restrict__ Wr,
    const float* __restrict__ Wvl, const float* __restrict__ Wvr,
    float* __restrict__ left, float* __restrict__ right,
    float* __restrict__ vleft, float* __restrict__ vright) {
  __shared__ float At[16][DIM];
  int bid = blockIdx.x;
  int side = bid >> 10;  // 16384 rows / 16 = 1024 tiles per side
  int mt = bid & 1023;
  int lane = threadIdx.x;

  for (int idx = lane; idx < 16 * DIM; idx += 32) {
    int row = idx >> 7, col = idx & 127;
    int r = mt * 16 + row;
    long src;
    if (side == 0) {
      int i = r >> 5, k = r & 31;
      src = ((long)i * L + anchor[k]) * DIM + col;
    } else {
      int k = r >> 9, j = r & (L - 1);
      src = ((long)anchor[k] * L + j) * DIM + col;
    }
    At[row][col] = pair[src];
  }
  __syncthreads();

  const float* W0 = side ? Wr : Wl;
  const float* W1 = side ? Wvr : Wvl;
  float* O0 = side ? right : left;
  float* O1 = side ? vright : vleft;

#pragma unroll
  for (int nt = 0; nt < 4; ++nt) {
    v8f c0 = {};
    v8f c1 = {};
    for (int kb = 0; kb < DIM; kb += 4) {
      v2f a = fragA(&At[0][0], DIM, kb, lane);
      c0 = wmma4(a, fragB(W0, ADIM, kb, nt * 16, lane), c0);
      c1 = wmma4(a, fragB(W1, ADIM, kb, nt * 16, lane), c1);
    }
    storeD(O0 + (long)mt * 16 * ADIM + nt * 16, ADIM, c0, lane);
    storeD(O1 + (long)mt * 16 * ADIM + nt * 16, ADIM, c1, lane);
  }
}

// -------- Kernel 2: fused attention over one (i, 16-wide j-tile) --------
// 128 threads = 4 waves. ~608 WMMAs per block. Single pass over pair_repr.

__global__ __launch_bounds__(128) void fused_attn_kernel(
    const float* __restrict__ pair, const float* __restrict__ td,
    const int* __restrict__ anchor,
    const float* __restrict__ Wq, const float* __restrict__ Wo,
    const float* __restrict__ Wg, const float* __restrict__ bg,
    const float* __restrict__ left, const float* __restrict__ right,
    const float* __restrict__ vleft, const float* __restrict__ vright,
    const float* __restrict__ gate_p, float* __restrict__ out) {
  __shared__ float P[16][DIM];     // pair rows
  __shared__ float Q[16][ADIM];    // q projection
  __shared__ float G[16][DIM];     // sigmoid gate
  __shared__ float S1[16][KANCH];  // q . left^T
  __shared__ float S2[16][KANCH];  // q . right (j-diagonal term)
  __shared__ float AT[16][KANCH];  // attn
  __shared__ float U[16][ADIM];    // update pre-Wo
  __shared__ float tl[KANCH];
  __shared__ float tr[16][KANCH];
  __shared__ float tdij[16];

  int i = blockIdx.x >> 5;  // 32 j-tiles per row i
  int jt = blockIdx.x & 31;
  int j0 = jt * 16;
  int tid = threadIdx.x;
  int wave = tid >> 5;  // uniform within a wave
  int lane = tid & 31;
  float gate = gate_p[0];

  for (int idx = tid; idx < 16 * DIM; idx += 128) {
    int r = idx >> 7, c = idx & 127;
    P[r][c] = pair[((long)i * L + j0 + r) * DIM + c];
  }
  if (tid < KANCH) tl[tid] = td[(long)i * L + anchor[tid]];
  for (int idx = tid; idx < 16 * KANCH; idx += 128) {
    int r = idx >> 5, k = idx & 31;
    tr[r][k] = td[(long)anchor[k] * L + j0 + r];
  }
  if (tid < 16) tdij[tid] = td[(long)i * L + j0 + tid];
  __syncthreads();

  // Stage 1: Q = P @ Wq (wave -> ntile), G = sigmoid(P @ Wg + bg) (wave -> 2 ntiles)
  {
    int nt = wave;
    v8f cq = {};
    v8f cg0 = {};
    v8f cg1 = {};
    for (int kb = 0; kb < DIM; kb += 4) {
      v2f a = fragA(&P[0][0], DIM, kb, lane);
      cq = wmma4(a, fragB(Wq, ADIM, kb, nt * 16, lane), cq);
      cg0 = wmma4(a, fragB(Wg, DIM, kb, nt * 16, lane), cg0);
      cg1 = wmma4(a, fragB(Wg, DIM, kb, (nt + 4) * 16, lane), cg1);
    }
    storeD(&Q[0][nt * 16], ADIM, cq, lane);
    int n = lane & 15, mb = (lane >> 4) << 3;
#pragma unroll
    for (int r = 0; r < 8; ++r) {
      float x0 = cg0[r] + bg[nt * 16 + n];
      float x1 = cg1[r] + bg[(nt + 4) * 16 + n];
      G[mb + r][nt * 16 + n] = 1.f / (1.f + __expf(-x0));
      G[mb + r][(nt + 4) * 16 + n] = 1.f / (1.f + __expf(-x1));
    }
  }
  __syncthreads();

  // Stage 2: S1[j,k] = sum_a Q[j,a]*left[i,k,a] via WMMA (B = left^T).
  // All waves run the WMMA (EXEC stays all-1s); only waves 0,1 store.
  {
    int nt = wave & 1;
    v8f c = {};
    const float* Lrow = left + (long)i * KANCH * ADIM;  // [k][a]
    for (int kb = 0; kb < ADIM; kb += 4)
      c = wmma4(fragA(&Q[0][0], ADIM, kb, lane), fragBT(Lrow, ADIM, kb, nt * 16, lane), c);
    if (wave < 2) storeD(&S1[0][nt * 16], KANCH, c, lane);
  }
  // S2[j,k] = sum_a Q[j,a]*right[k, j0+j, a]  (j-dependent B: batched matvec, VALU)
  for (int p = tid; p < 16 * KANCH; p += 128) {
    int j = p >> 5, k = p & 31;
    const float* rrow = right + ((long)k * L + j0 + j) * ADIM;
    float s = 0.f;
#pragma unroll 8
    for (int a = 0; a < ADIM; ++a) s += Q[j][a] * rrow[a];
    S2[j][k] = s;
  }
  __syncthreads();

  // Stage 3: bias + softmax over K=32 (one thread per row j)
  if (tid < 16) {
    int j = tid;
    float sc[KANCH];
    float mx = -1e30f;
#pragma unroll
    for (int k = 0; k < KANCH; ++k) {
      float b = tl[k] + tr[j][k] - tdij[j];
      b = -fabsf(b) * 0.25f;  // / SIGMA(=4)
      float v = (S1[j][k] + S2[j][k]) * 0.125f + gate * b;  // 1/sqrt(64)
      sc[k] = v;
      mx = fmaxf(mx, v);
    }
    float sum = 0.f;
#pragma unroll
    for (int k = 0; k < KANCH; ++k) {
      float e = __expf(sc[k] - mx);
      sc[k] = e;
      sum += e;
    }
    float inv = 1.f / sum;
#pragma unroll
    for (int k = 0; k < KANCH; ++k) AT[j][k] = sc[k] * inv;
  }
  __syncthreads();

  // Stage 4: U = attn @ v_left[i]  (WMMA, wave -> ntile of 4)
  {
    int nt = wave;
    v8f c = {};
    const float* VL = vleft + (long)i * KANCH * ADIM;  // [k][a]
    for (int kb = 0; kb < KANCH; kb += 4)
      c = wmma4(fragA(&AT[0][0], KANCH, kb, lane), fragB(VL, ADIM, kb, nt * 16, lane), c);
    storeD(&U[0][nt * 16], ADIM, c, lane);
  }
  __syncthreads();

  // Stage 5: U += attn @ v_right (j-dependent, VALU)
  for (int p = tid; p < 16 * ADIM; p += 128) {
    int j = p >> 6, a = p & 63;
    float s = 0.f;
#pragma unroll
    for (int k = 0; k < KANCH; ++k)
      s += AT[j][k] * vright[((long)k * L + j0 + j) * ADIM + a];
    U[j][a] += s;
  }
  __syncthreads();

  // Stage 6: Y = U @ Wo ; out = P + G * Y  (wave -> 2 ntiles of 8)
#pragma unroll
  for (int half = 0; half < 2; ++half) {
    int nt = wave + half * 4;
    v8f c = {};
    for (int kb = 0; kb < ADIM; kb += 4)
      c = wmma4(fragA(&U[0][0], ADIM, kb, lane), fragB(Wo, DIM, kb, nt * 16, lane), c);
    int n = lane & 15, mb = (lane >> 4) << 3;
#pragma unroll
    for (int r = 0; r < 8; ++r) {
      int row = mb + r, col = nt * 16 + n;
      out[((long)i * L + j0 + row) * DIM + col] = P[row][col] + G[row][col] * c[r];
    }
  }
}

extern "C" void kernel_launch(void* const* d_in, const int* in_sizes, int n_in,
                              void* d_out, int out_size, void* d_ws, size_t ws_size,
                              hipStream_t stream) {
  const float* pair = (const float*)d_in[0];
  const float* td = (const float*)d_in[1];
  const float* qual = (const float*)d_in[2];
  const float* Wq = (const float*)d_in[3];
  const float* Wl = (const float*)d_in[4];
  const float* Wr = (const float*)d_in[5];
  const float* Wvl = (const float*)d_in[6];
  const float* Wvr = (const float*)d_in[7];
  const float* Wo = (const float*)d_in[8];
  const float* Wg = (const float*)d_in[9];
  const float* bg = (const float*)d_in[10];
  const float* T1 = (const float*)d_in[11];
  const float* Tb1 = (const float*)d_in[12];
  const float* T2 = (const float*)d_in[13];
  const float* Tb2 = (const float*)d_in[14];
  const int* anchor = (const int*)d_in[15];
  float* out = (float*)d_out;

  float* ws = (float*)d_ws;
  const long NP = (long)L * KANCH * ADIM;  // 1,048,576 floats per projection
  float* gate = ws;                        // 16 floats (padded)
  float* left = ws + 16;
  float* right = left + NP;
  float* vleft = right + NP;
  float* vright = vleft + NP;

  gate_kernel<<<1, 256, 0, stream>>>(td, qual, T1, Tb1, T2, Tb2, gate);
  anchor_proj_kernel<<<2048, 32, 0, stream>>>(pair, anchor, Wl, Wr, Wvl, Wvr,
                                              left, right, vleft, vright);
  fused_attn_kernel<<<L * (L / 16), 128, 0, stream>>>(
      pair, td, anchor, Wq, Wo, Wg, bg, left, right, vleft, vright, gate, out);
}